// OptimizedMatrixFactorizationModel_86517821216463
// MI455X (gfx1250) — compile-verified
//
#include <hip/hip_runtime.h>
#include <hip/hip_bf16.h>

typedef __attribute__((ext_vector_type(2))) float v2f;
typedef __attribute__((ext_vector_type(8))) float v8f;

#define DIM 64
#define CHUNK 128           // tokens per wave in segment-sum
#define WAVES_PER_BLOCK 8   // seg-sum: 256 threads
#define PRED_WAVES 4        // pred: 128 threads, 1 tile of 16 samples per wave
#define ROWSTRIDE 68        // padded LDS row stride (floats) -> conflict-free

// ---------------------------------------------------------------------------
// Kernel 1: gather rows, weight by val, register-accumulate runs of equal seg
// (seg is sorted), flush with float atomics. One wave handles CHUNK tokens;
// 32 lanes each own a float2 slice of the 64-dim row (one 256B coalesced load
// per token).
// ---------------------------------------------------------------------------
__global__ void seg_accum_kernel(const float* __restrict__ table,
                                 const int*   __restrict__ idx,
                                 const float* __restrict__ val,
                                 const int*   __restrict__ seg,
                                 float*       __restrict__ acc_out,
                                 int T)
{
    const int lane = threadIdx.x & 31;
    const int gwave = blockIdx.x * (blockDim.x >> 5) + (threadIdx.x >> 5);
    const int base = gwave * CHUNK;
    if (base >= T) return;

    int   cur = -1;
    float ax = 0.f, ay = 0.f;

    for (int sub = 0; sub < CHUNK; sub += 32) {
        const int t0 = base + sub;
        if (t0 >= T) break;
        const int t = t0 + lane;
        int   myseg = 0, myidx = 0;
        float myval = 0.f;
        if (t < T) {
            myseg = seg[t];
            myidx = idx[t];
            myval = val[t];
        }
        // gfx1250 prefetch of the 32 upcoming rows (one row per lane)
        __builtin_prefetch(table + (size_t)myidx * DIM, 0, 3);

        const int n = min(32, T - t0);
        for (int j = 0; j < n; j++) {
            const int   s  = __shfl(myseg, j);
            const int   id = __shfl(myidx, j);
            const float v  = __shfl(myval, j);
            if (s != cur) {                       // wave-uniform branch
                if (cur >= 0) {
                    float* p = acc_out + (size_t)cur * DIM + 2 * lane;
                    __hip_atomic_fetch_add(p,     ax, __ATOMIC_RELAXED, __HIP_MEMORY_SCOPE_AGENT);
                    __hip_atomic_fetch_add(p + 1, ay, __ATOMIC_RELAXED, __HIP_MEMORY_SCOPE_AGENT);
                }
                cur = s; ax = 0.f; ay = 0.f;
            }
            const float2 e = *(const float2*)(table + (size_t)id * DIM + 2 * lane);
            ax = fmaf(e.x, v, ax);
            ay = fmaf(e.y, v, ay);
        }
    }
    if (cur >= 0) {
        float* p = acc_out + (size_t)cur * DIM + 2 * lane;
        __hip_atomic_fetch_add(p,     ax, __ATOMIC_RELAXED, __HIP_MEMORY_SCOPE_AGENT);
        __hip_atomic_fetch_add(p + 1, ay, __ATOMIC_RELAXED, __HIP_MEMORY_SCOPE_AGENT);
    }
}

// ---------------------------------------------------------------------------
// Kernel 2: per-tile (16 samples) prediction via WMMA.
//   u = user_emb[uid] + u_acc ;  v = item_emb[iid] + i_acc   (staged in LDS)
//   pred = diag(U * V^T) + biases, with U:16x64, V^T:64x16 done as
//   16 chained v_wmma_f32_16x16x4_f32 over K.
// ---------------------------------------------------------------------------
__global__ void mf_pred_kernel(const int*   __restrict__ user_ids,
                               const int*   __restrict__ item_ids,
                               const float* __restrict__ user_emb,
                               const float* __restrict__ item_emb,
                               const float* __restrict__ user_bias,
                               const float* __restrict__ item_bias,
                               const float* __restrict__ g_bias,
                               const float* __restrict__ u_acc,
                               const float* __restrict__ i_acc,
                               float*       __restrict__ out,
                               int B)
{
    // per-wave LDS: u[16][68], v[16][68], pred[16] (+pad)
    __shared__ float smem[PRED_WAVES * (2 * 16 * ROWSTRIDE + 32)];
    const int lane = threadIdx.x & 31;
    const int wv   = threadIdx.x >> 5;

    const int tiles = (B + 15) >> 4;
    int tile = blockIdx.x * PRED_WAVES + wv;
    if (tile >= tiles) tile = tiles - 1;       // clamp: redundant but uniform

    float* u_l = smem + wv * (2 * 16 * ROWSTRIDE + 32);
    float* v_l = u_l + 16 * ROWSTRIDE;
    float* p_l = v_l + 16 * ROWSTRIDE;

    const int s0 = tile * 16;

    // ---- stage u, v tiles into LDS (coalesced float2 per lane) ----
    #pragma unroll 4
    for (int m = 0; m < 16; m++) {
        int s = s0 + m;  if (s >= B) s = B - 1;
        const int uid = user_ids[s];
        const int iid = item_ids[s];
        const float2 ue = *(const float2*)(user_emb + (size_t)uid * DIM + 2 * lane);
        const float2 ua = *(const float2*)(u_acc    + (size_t)s   * DIM + 2 * lane);
        const float2 ie = *(const float2*)(item_emb + (size_t)iid * DIM + 2 * lane);
        const float2 ia = *(const float2*)(i_acc    + (size_t)s   * DIM + 2 * lane);
        u_l[m * ROWSTRIDE + 2 * lane]     = ue.x + ua.x;
        u_l[m * ROWSTRIDE + 2 * lane + 1] = ue.y + ua.y;
        v_l[m * ROWSTRIDE + 2 * lane]     = ie.x + ia.x;
        v_l[m * ROWSTRIDE + 2 * lane + 1] = ie.y + ia.y;
    }
    __syncthreads();

    // ---- D = U * V^T via 16 x wmma_f32_16x16x4_f32 (full EXEC) ----
    // A 16x4 f32: lane L holds A[L&15][koff], A[L&15][koff+1], koff = (L>>4)*2
    // B 4x16 f32: lane L holds B[koff][L&15], B[koff+1][L&15] = v[L&15][koff..]
    const int mrow = lane & 15;
    const int koff = (lane >> 4) << 1;
    v8f c = {};
    #pragma unroll
    for (int kk = 0; kk < 16; kk++) {
        v2f a = *(const v2f*)&u_l[mrow * ROWSTRIDE + kk * 4 + koff];
        v2f b = *(const v2f*)&v_l[mrow * ROWSTRIDE + kk * 4 + koff];
        c = __builtin_amdgcn_wmma_f32_16x16x4_f32(false, a, false, b,
                                                  (short)0, c, false, false);
    }

    // ---- extract diag: sample m<8 at lane m (c[m]); m>=8 at lane m+16 (c[m-8])
    float x = c[0];
    #pragma unroll
    for (int r = 1; r < 8; r++)
        if ((lane & 7) == r) x = c[r];
    if (lane < 8)   p_l[lane] = x;          // samples 0..7
    if (lane >= 24) p_l[lane - 16] = x;     // samples 8..15
    __syncthreads();

    int s = s0 + (lane & 15);
    const int sc = (s >= B) ? (B - 1) : s;
    const float p = p_l[lane & 15] + user_bias[user_ids[sc]]
                                   + item_bias[item_ids[sc]] + g_bias[0];
    if (lane < 16 && s < B) out[s] = p;
}

extern "C" void kernel_launch(void* const* d_in, const int* in_sizes, int n_in,
                              void* d_out, int out_size, void* d_ws, size_t ws_size,
                              hipStream_t stream) {
    const int*   user_ids      = (const int*)  d_in[0];
    const int*   item_ids      = (const int*)  d_in[1];
    const int*   user_feat_idx = (const int*)  d_in[2];
    const float* user_feat_val = (const float*)d_in[3];
    const int*   user_seg      = (const int*)  d_in[4];
    const int*   item_feat_idx = (const int*)  d_in[5];
    const float* item_feat_val = (const float*)d_in[6];
    const int*   item_seg      = (const int*)  d_in[7];
    const float* user_emb      = (const float*)d_in[8];
    const float* item_emb      = (const float*)d_in[9];
    const float* user_feat_emb = (const float*)d_in[10];
    const float* item_feat_emb = (const float*)d_in[11];
    const float* user_bias     = (const float*)d_in[12];
    const float* item_bias     = (const float*)d_in[13];
    const float* global_bias   = (const float*)d_in[14];

    const int B   = in_sizes[0];
    const int T_U = in_sizes[2];
    const int T_I = in_sizes[5];

    float* u_acc = (float*)d_ws;
    float* i_acc = u_acc + (size_t)B * DIM;

    // zero the two [B, DIM] accumulators (graph-capture safe)
    hipMemsetAsync(d_ws, 0, (size_t)2 * B * DIM * sizeof(float), stream);

    // segment-sum: one wave per CHUNK tokens
    {
        const int nChunksU = (T_U + CHUNK - 1) / CHUNK;
        const int blocksU  = (nChunksU + WAVES_PER_BLOCK - 1) / WAVES_PER_BLOCK;
        seg_accum_kernel<<<blocksU, 32 * WAVES_PER_BLOCK, 0, stream>>>(
            user_feat_emb, user_feat_idx, user_feat_val, user_seg, u_acc, T_U);

        const int nChunksI = (T_I + CHUNK - 1) / CHUNK;
        const int blocksI  = (nChunksI + WAVES_PER_BLOCK - 1) / WAVES_PER_BLOCK;
        seg_accum_kernel<<<blocksI, 32 * WAVES_PER_BLOCK, 0, stream>>>(
            item_feat_emb, item_feat_idx, item_feat_val, item_seg, i_acc, T_I);
    }

    // prediction: one 16-sample tile per wave
    {
        const int tiles  = (B + 15) / 16;
        const int blocks = (tiles + PRED_WAVES - 1) / PRED_WAVES;
        mf_pred_kernel<<<blocks, 32 * PRED_WAVES, 0, stream>>>(
            user_ids, item_ids, user_emb, item_emb,
            user_bias, item_bias, global_bias,
            u_acc, i_acc, (float*)d_out, B);
    }
}